// NewCrossEntropy_18674517803017
// MI455X (gfx1250) — compile-verified
//
#include <hip/hip_runtime.h>

// NewCrossEntropy for MI455X (gfx1250, wave32).
// Streaming one-pass online log-sum-exp, bandwidth-bound (80 MB read, ~3.4 us roofline).
// 4x-unrolled NT b128 loads for per-wave MLP; deterministic two-kernel tree reduction.

typedef __attribute__((ext_vector_type(4))) float f4;

#define NBLK 1024   // blocks per row
#define TPB  256    // 8 wave32s per block

// Combine two (max, scaled-sum) logsumexp partials. s==0 means "empty".
__device__ __forceinline__ void lse_combine(float& m, float& s, float mo, float so) {
    if (so == 0.0f) return;
    if (s == 0.0f) { m = mo; s = so; return; }
    float M = fmaxf(m, mo);
    s = s * __expf(m - M) + so * __expf(mo - M);
    m = M;
}

// Online update with a single scalar element.
__device__ __forceinline__ void lse_push(float& m, float& s, float x) {
    if (x <= m) {
        s += __expf(x - m);
    } else {
        s = s * __expf(m - x) + 1.0f;
        m = x;
    }
}

// Online update with 4 elements; common case (no new max) costs 4 exps, no rescale.
__device__ __forceinline__ void lse_push4(float& m, float& s, f4 v) {
    float vm = fmaxf(fmaxf(v.x, v.y), fmaxf(v.z, v.w));
    if (vm <= m) {
        s += __expf(v.x - m) + __expf(v.y - m) + __expf(v.z - m) + __expf(v.w - m);
    } else {
        // exp(-inf - vm) == 0 handles the first iteration (m = -inf, s = 0)
        float r = __expf(m - vm);
        s = s * r + __expf(v.x - vm) + __expf(v.y - vm)
                  + __expf(v.z - vm) + __expf(v.w - vm);
        m = vm;
    }
}

// Kernel 1: per-block (m, s) logsumexp partials. grid = (NBLK, 2 rows).
__global__ void ce_partial_kernel(const float* __restrict__ outputs,
                                  float2* __restrict__ partials, int C) {
    const int row = blockIdx.y;
    const float* __restrict__ rowp = outputs + (size_t)row * (size_t)C;

    const int tid      = blockIdx.x * blockDim.x + threadIdx.x;
    const int nthreads = gridDim.x * blockDim.x;   // threads per row

    float m = -INFINITY;
    float s = 0.0f;

    const int nvec = C >> 2;                 // float4 count
    const f4* __restrict__ vp = (const f4*)rowp;

    // Batched grid-stride: 4 outstanding NT b128 loads per wave per iteration.
    int i = tid;
    for (; i + 3 * nthreads < nvec; i += 4 * nthreads) {
        // speculative prefetch one full batch ahead (global_prefetch_b8; OOB dropped)
        __builtin_prefetch((const void*)(vp + i + 4 * nthreads), 0, 0);

        f4 v0 = __builtin_nontemporal_load(vp + i);
        f4 v1 = __builtin_nontemporal_load(vp + i + nthreads);
        f4 v2 = __builtin_nontemporal_load(vp + i + 2 * nthreads);
        f4 v3 = __builtin_nontemporal_load(vp + i + 3 * nthreads);

        lse_push4(m, s, v0);
        lse_push4(m, s, v1);
        lse_push4(m, s, v2);
        lse_push4(m, s, v3);
    }
    // vector remainder
    for (; i < nvec; i += nthreads) {
        f4 v = __builtin_nontemporal_load(vp + i);
        lse_push4(m, s, v);
    }
    // generic scalar tail (empty for C = 10,000,000)
    for (int j = (nvec << 2) + tid; j < C; j += nthreads) {
        lse_push(m, s, rowp[j]);
    }

    // wave32 butterfly reduction
    #pragma unroll
    for (int off = 16; off >= 1; off >>= 1) {
        float mo = __shfl_xor(m, off, 32);
        float so = __shfl_xor(s, off, 32);
        lse_combine(m, s, mo, so);
    }

    // cross-wave reduction through LDS (8 waves)
    __shared__ float sm[TPB / 32];
    __shared__ float ss[TPB / 32];
    const int lane = threadIdx.x & 31;
    const int wave = threadIdx.x >> 5;
    if (lane == 0) { sm[wave] = m; ss[wave] = s; }
    __syncthreads();
    if (threadIdx.x == 0) {
        #pragma unroll
        for (int w = 1; w < TPB / 32; ++w) lse_combine(m, s, sm[w], ss[w]);
        partials[(size_t)row * gridDim.x + blockIdx.x] = make_float2(m, s);
    }
}

// Kernel 2: single block combines partials, gathers target logits, writes loss.
__global__ void ce_final_kernel(const float* __restrict__ outputs,
                                const long long* __restrict__ labels,
                                const float* __restrict__ a,
                                const float2* __restrict__ partials,
                                float* __restrict__ out, int C, int nblk) {
    __shared__ float sm[TPB / 32];
    __shared__ float ss[TPB / 32];
    __shared__ float rowM[2];
    __shared__ float rowS[2];

    const int tid  = threadIdx.x;
    const int lane = tid & 31;
    const int wave = tid >> 5;

    for (int r = 0; r < 2; ++r) {
        float m = -INFINITY;
        float s = 0.0f;
        for (int i = tid; i < nblk; i += TPB) {
            float2 p = partials[(size_t)r * nblk + i];
            lse_combine(m, s, p.x, p.y);
        }
        #pragma unroll
        for (int off = 16; off >= 1; off >>= 1) {
            float mo = __shfl_xor(m, off, 32);
            float so = __shfl_xor(s, off, 32);
            lse_combine(m, s, mo, so);
        }
        if (lane == 0) { sm[wave] = m; ss[wave] = s; }
        __syncthreads();
        if (tid == 0) {
            #pragma unroll
            for (int w = 1; w < TPB / 32; ++w) lse_combine(m, s, sm[w], ss[w]);
            rowM[r] = m; rowS[r] = s;
        }
        __syncthreads();
    }

    if (tid == 0) {
        float loss = 0.0f;
        for (int r = 0; r < 2; ++r) {
            long long lbl = labels[r];
            float xt = outputs[(size_t)r * (size_t)C + (size_t)lbl];
            float nll = rowM[r] + logf(rowS[r]) - xt;   // max + log(sumexp) - x[label]
            loss += nll * a[r];
        }
        out[0] = loss;
    }
}

extern "C" void kernel_launch(void* const* d_in, const int* in_sizes, int n_in,
                              void* d_out, int out_size, void* d_ws, size_t ws_size,
                              hipStream_t stream) {
    const float*     outputs = (const float*)d_in[0];      // [2, C] fp32
    const long long* labels  = (const long long*)d_in[1];  // [2] int64
    const float*     aw      = (const float*)d_in[2];      // [2] fp32

    const int C = in_sizes[0] / 2;

    float2* partials = (float2*)d_ws;   // 2 * NBLK * 8 B = 16 KB

    dim3 grid(NBLK, 2);
    ce_partial_kernel<<<grid, TPB, 0, stream>>>(outputs, partials, C);
    ce_final_kernel<<<1, TPB, 0, stream>>>(outputs, labels, aw, partials,
                                           (float*)d_out, C, NBLK);
}